// KManifoldClusterModel_49555332661828
// MI455X (gfx1250) — compile-verified
//
#include <hip/hip_runtime.h>
#include <hip/hip_bf16.h>

typedef __attribute__((ext_vector_type(2))) float v2f;
typedef __attribute__((ext_vector_type(4))) float v4f;
typedef __attribute__((ext_vector_type(8))) float v8f;

// Problem constants (fixed by the reference): N=200000, n=64, d=16, D=512, B=4096
#define NJ   64      // clusters (innermost output dim)
#define ND   16      // d (contraction)
#define NK   512     // D
#define NB   4096    // batch
#define BT   16      // b-tile per workgroup
#define KCH  16      // k-chunk computed per pass (one WMMA tile wide)

// LDS padded strides (floats)
#define VJ_STRIDE 257            // v_lds[j*257 + d*16 + b]
#define UK_STRIDE 17             // u_lds[j*272 + k*17 + d]
#define UJ_STRIDE (16*UK_STRIDE) // 272
#define OM_STRIDE 1040           // o_lds[m*1040 + nk*65 + j]
#define ON_STRIDE 65

__launch_bounds__(256, 1)
__global__ void kmfc_kernel(const int* __restrict__ ii,
                            const float* __restrict__ C,
                            const float* __restrict__ V,
                            const float* __restrict__ U,
                            float* __restrict__ out_x,
                            float* __restrict__ out_c,
                            float* __restrict__ out_v)
{
    __shared__ float v_lds[NJ * VJ_STRIDE];   // 64*257*4  = 65,792 B
    __shared__ float u_lds[NJ * UJ_STRIDE];   // 64*272*4  = 69,632 B
    __shared__ float o_lds[BT * OM_STRIDE];   // 16*1040*4 = 66,560 B
    __shared__ int   rows[BT];

    const int tid = threadIdx.x;
    const int bt  = blockIdx.x >> 3;    // b-tile index: 0..255
    const int kg  = blockIdx.x & 7;     // k-group (64 k each): 0..7
    const int b0  = bt * BT;

    if (tid < BT) rows[tid] = ii[b0 + tid];
    __syncthreads();

    // ---- Gather v rows into LDS (transposed to [j][d][b]); emit v output once ----
    for (int i = tid; i < BT * ND * NJ; i += 256) {
        const int b = i >> 10;          // /(16*64)
        const int d = (i >> 6) & 15;
        const int j = i & 63;
        const float g = V[(size_t)rows[b] * (ND * NJ) + d * NJ + j];
        v_lds[j * VJ_STRIDE + d * 16 + b] = g;
        if (kg == 0)
            __builtin_nontemporal_store(g, &out_v[(size_t)(b0 + b) * (ND * NJ) + d * NJ + j]);
    }

    // ---- c output: squared-normalized soft assignment, clamped (kg==0 only) ----
    if (kg == 0 && tid < BT) {
        const size_t row = (size_t)rows[tid] * NJ;
        float s = 0.f;
        for (int j = 0; j < NJ; ++j) { const float x = C[row + j]; s += x * x; }
        const float inv = 1.0f / (s + 1e-8f);
        for (int j = 0; j < NJ; ++j) {
            const float x = C[row + j];
            float y = x * x * inv;
            y = fminf(fmaxf(y, 0.1f / 64.0f), 1.0f);
            __builtin_nontemporal_store(y, &out_c[(size_t)(b0 + tid) * NJ + j]);
        }
    }

    const int wave = tid >> 5;
    const int lane = tid & 31;
    const int half = lane >> 4;     // K-group half for A/B operands
    const int ln   = lane & 15;     // M (A) / N (B,C,D) index

    for (int kc = 0; kc < 4; ++kc) {
        const int kbase = kg * 64 + kc * KCH;
        __syncthreads();   // u_lds / o_lds reuse from previous chunk; v_lds ready (kc==0)

        // ---- Load U chunk [64 j][16 k][16 d] into padded LDS ----
        for (int i = tid; i < NJ * KCH * ND; i += 256) {
            const int j  = i >> 8;
            const int kk = (i >> 4) & 15;
            const int d  = i & 15;
            u_lds[j * UJ_STRIDE + kk * UK_STRIDE + d] =
                U[(size_t)j * (NK * ND) + (size_t)(kbase + kk) * ND + d];
        }
        __syncthreads();

        // ---- WMMA: each of 8 waves computes 8 clusters' 16x16 tiles (K=16 via 4 chained x4) ----
        for (int jj = 0; jj < 8; ++jj) {
            const int j = wave * 8 + jj;
            v8f acc = {};
            const float* vj = &v_lds[j * VJ_STRIDE];
            const float* uj = &u_lds[j * UJ_STRIDE];
#pragma unroll
            for (int dd = 0; dd < 4; ++dd) {
                const int d0 = dd * 4 + half * 2;           // K index this lane-half supplies
                v2f a, bm;
                a.x  = vj[d0 * 16 + ln];                    // A[m=ln][k=d0]
                a.y  = vj[(d0 + 1) * 16 + ln];              // A[m=ln][k=d0+1]
                bm.x = uj[ln * UK_STRIDE + d0];             // B[k=d0][n=ln] = U[j][kbase+ln][d0]
                bm.y = uj[ln * UK_STRIDE + d0 + 1];
                acc = __builtin_amdgcn_wmma_f32_16x16x4_f32(
                        false, a, false, bm, (short)0, acc, false, false);
            }
            // D layout: VGPR r -> M=r+8*half, lane%16 -> N. Stage transposed to [m][k][j].
#pragma unroll
            for (int r = 0; r < 8; ++r)
                o_lds[(r + half * 8) * OM_STRIDE + ln * ON_STRIDE + j] = acc[r];
        }
        __syncthreads();

        // ---- Coalesced NT stores: each (m,nk) row is 64 contiguous floats in x_ ----
        for (int rnd = 0; rnd < 16; ++rnd) {
            const int p  = (tid >> 4) + rnd * 16;   // (m,nk) pair: 0..255
            const int m  = p >> 4;
            const int nk = p & 15;
            const int j4 = (tid & 15) * 4;
            const float* src = &o_lds[m * OM_STRIDE + nk * ON_STRIDE + j4];
            v4f val = { src[0], src[1], src[2], src[3] };
            __builtin_nontemporal_store(val,
                (v4f*)&out_x[(size_t)(b0 + m) * (NK * NJ) + (size_t)(kbase + nk) * NJ + j4]);
        }
    }
}

extern "C" void kernel_launch(void* const* d_in, const int* in_sizes, int n_in,
                              void* d_out, int out_size, void* d_ws, size_t ws_size,
                              hipStream_t stream) {
    const int*   ii = (const int*)  d_in[0];
    const float* C  = (const float*)d_in[1];
    const float* V  = (const float*)d_in[2];
    const float* U  = (const float*)d_in[3];

    float* out_x = (float*)d_out;                                  // [4096,512,64]
    float* out_c = out_x + (size_t)NB * NK * NJ;                   // [4096,64]
    float* out_v = out_c + (size_t)NB * NJ;                        // [4096,16,64]

    const int blocks = (NB / BT) * 8;   // 256 b-tiles x 8 k-groups = 2048
    kmfc_kernel<<<blocks, 256, 0, stream>>>(ii, C, V, U, out_x, out_c, out_v);
}